// Retina_DIBA_21517786153595
// MI455X (gfx1250) — compile-verified
//
#include <hip/hip_runtime.h>

// ---------------- constants ----------------
#define SCORE_TOTAL 10927917
#define L0_END 8190000
#define L1_END 10237500
#define L2_END 10749375
#define L3_END 10887786
#define THR_LOGIT (-2.9444389791664403f)   // log(0.05/0.95)
#define BBOX_CLIPF 4.135166556742356f      // log(1000/16)
#define IMGF 800.0f
#define TOPK 1000
#define NCAND 5000
#define NCAND_PAD 5024
#define BND_CAP 8192
#define DETS 300
#define MASK_STRIDE 160
#define NJW 157      // 5024/32 column-words
#define NTI 314      // 5024/16 row tiles
#define K1_CHUNK 4096
#define N_FULL_CHUNKS 2667   // SCORE_TOTAL / K1_CHUNK

#define M_CUT 0
#define M_NEED 5
#define M_ABOVE 10
#define M_OUT 16
#define M_BND 24

typedef unsigned u32;
typedef unsigned long long u64;
typedef float v2f __attribute__((ext_vector_type(2)));
typedef float v8f __attribute__((ext_vector_type(8)));

// ---------------- CDNA5 feature probes ----------------
#if defined(__AMDGCN__) && __has_builtin(__builtin_amdgcn_global_load_async_to_lds_b128)
#define HAVE_ASYNC_LDS 1
#else
#define HAVE_ASYNC_LDS 0
#endif

#if defined(__AMDGCN__) && __has_builtin(__builtin_amdgcn_wmma_f32_16x16x4_f32)
#define HAVE_WMMA4 1
#else
#define HAVE_WMMA4 0
#endif

__device__ __forceinline__ u32 ballot32(bool p) {
#if defined(__AMDGCN__) && __has_builtin(__builtin_amdgcn_ballot_w32)
  return __builtin_amdgcn_ballot_w32(p);
#else
  return (u32)__ballot(p);
#endif
}

#if HAVE_ASYNC_LDS
// Builtin signature (from hipcc diagnostic): param0 = int4 (vector_size 16) pointer in
// AS1 ("__device__"), param1 = int4 pointer in AS3 (LDS), then imm offset + cpol.
typedef int v4i_ __attribute__((vector_size(16)));
typedef __attribute__((address_space(1))) v4i_ GlobV4;
typedef __attribute__((address_space(3))) v4i_ LdsV4;
__device__ __forceinline__ void asyncLoad16(const void* g, void* l) {
  __builtin_amdgcn_global_load_async_to_lds_b128((GlobV4*)g, (LdsV4*)l, 0, 0);
}
__device__ __forceinline__ void asyncWait4() {
#if __has_builtin(__builtin_amdgcn_s_wait_asynccnt)
  __builtin_amdgcn_s_wait_asynccnt(4);
#else
  asm volatile("s_wait_asynccnt 4" ::: "memory");
#endif
}
__device__ __forceinline__ void asyncWait0() {
#if __has_builtin(__builtin_amdgcn_s_wait_asynccnt)
  __builtin_amdgcn_s_wait_asynccnt(0);
#else
  asm volatile("s_wait_asynccnt 0" ::: "memory");
#endif
}
#else
__device__ __forceinline__ void asyncLoad16(const void* g, void* l) {
  *(float4*)l = *(const float4*)g;
}
__device__ __forceinline__ void asyncWait4() {}
__device__ __forceinline__ void asyncWait0() {}
#endif

// ---------------- helpers ----------------
__device__ __forceinline__ int lvlOf(int gi) {
  return (gi < L0_END) ? 0 : (gi < L1_END) ? 1 : (gi < L2_END) ? 2 : (gi < L3_END) ? 3 : 4;
}
__device__ __forceinline__ int scoreOff(int l) {
  const int off[5] = {0, L0_END, L1_END, L2_END, L3_END};
  return off[l];
}
__device__ __forceinline__ int anchOff(int l) {
  const int off[5] = {0, 90000, 112500, 118125, 119646};
  return off[l];
}
__device__ __forceinline__ u32 orderKey(float f) {   // monotone float -> uint
  u32 u = __float_as_uint(f);
  return (u & 0x80000000u) ? ~u : (u | 0x80000000u);
}
__device__ __forceinline__ float keyToFloat(u32 k) {
  u32 u = (k & 0x80000000u) ? (k & 0x7FFFFFFFu) : ~k;
  return __uint_as_float(u);
}

// descending bitonic sort of n (pow2) u64 keys in LDS
__device__ __forceinline__ void bitonicDesc(u64* a, int n, int tid, int nthr) {
  for (int k = 2; k <= n; k <<= 1) {
    for (int j = k >> 1; j > 0; j >>= 1) {
      for (int i = tid; i < n; i += nthr) {
        int ix = i ^ j;
        if (ix > i) {
          u64 x = a[i], y = a[ix];
          bool desc = ((i & k) == 0);
          bool sw = desc ? (x < y) : (x > y);
          if (sw) { a[i] = y; a[ix] = x; }
        }
      }
      __syncthreads();
    }
  }
}

// ---------------- K0: init ----------------
__global__ void k_init(u32* hist, u32* meta) {
  int t = threadIdx.x;
  for (int i = t; i < 5120; i += 1024) hist[i] = 0u;
  if (t < 64) meta[t] = 0u;
}

// ---------------- K1: 1024-bin histogram per level (async LDS staged) ----------------
__device__ __forceinline__ void procScore(int gi, float f, u32* lh) {
  int lvl = lvlOf(gi);
  u32 key = 0u;
  if (f > THR_LOGIT) key = orderKey(f);
  atomicAdd(&lh[(lvl << 10) + (key >> 22)], 1u);
}

__device__ __forceinline__ void stageChunk(float* dst, const float* src, int tid) {
#pragma unroll
  for (int p = 0; p < 4; ++p) {
    int e = (p * 256 + tid) * 4;
    asyncLoad16(src + e, dst + e);
  }
}

__global__ void __launch_bounds__(256) k_hist(const float* __restrict__ logits, u32* __restrict__ hist) {
  __shared__ __align__(16) float buf[2][K1_CHUNK];
  __shared__ u32 lh[5 * 1024];
  const int tid = threadIdx.x;
  for (int i = tid; i < 5 * 1024; i += 256) lh[i] = 0u;
  __syncthreads();

  if (blockIdx.x == 0) { // ragged tail, direct loads (no OOB risk)
    for (int i = N_FULL_CHUNKS * K1_CHUNK + tid; i < SCORE_TOTAL; i += 256)
      procScore(i, logits[i], lh);
  }

  int cpb = (N_FULL_CHUNKS + gridDim.x - 1) / gridDim.x;
  int c0 = blockIdx.x * cpb;
  int c1 = min(c0 + cpb, N_FULL_CHUNKS);
  if (c0 < c1) {
    int cur = 0;
    stageChunk(buf[0], logits + (size_t)c0 * K1_CHUNK, tid);
    for (int c = c0; c < c1; ++c) {
      bool hasNext = (c + 1 < c1);
      if (hasNext) stageChunk(buf[1 - cur], logits + (size_t)(c + 1) * K1_CHUNK, tid);
      if (hasNext) asyncWait4(); else asyncWait0();
      __syncthreads();
      int base = c * K1_CHUNK;
      for (int e = tid; e < K1_CHUNK; e += 256)
        procScore(base + e, buf[cur][e], lh);
      __syncthreads();
      cur ^= 1;
    }
  }
  __syncthreads();
  for (int i = tid; i < 5 * 1024; i += 256) {
    u32 v = lh[i];
    if (v) atomicAdd(&hist[i], v);
  }
}

// ---------------- K2: find cutoff bin per level ----------------
__global__ void k_cutoff(const u32* __restrict__ hist, u32* meta) {
  int l = threadIdx.x;
  if (l >= 5) return;
  u32 acc = 0, cut = 0, above = 0;
  for (int b = 1023; b >= 0; --b) {
    u32 h = hist[l * 1024 + b];
    if (acc + h >= (u32)TOPK) { cut = (u32)b; above = acc; break; }
    acc += h;
    if (b == 0) { cut = 0u; above = acc - hist[l * 1024]; } // unreachable in practice
  }
  meta[M_CUT + l] = cut;
  meta[M_ABOVE + l] = above;
  meta[M_NEED + l] = (u32)TOPK - above;
}

// ---------------- K3: compaction ----------------
__device__ __forceinline__ void compactOne(int gi, float f, const u32* cuts, u32* meta,
                                           u32* topKey, u32* topIdx, u64* bnd) {
  int lvl = lvlOf(gi);
  u32 key = 0u;
  if (f > THR_LOGIT) key = orderKey(f);
  u32 bin = key >> 22;
  u32 cb = cuts[lvl];
  if (bin < cb) return;
  u32 idxL = (u32)(gi - scoreOff(lvl));
  if (bin > cb) {
    u32 pos = atomicAdd(&meta[M_OUT + lvl], 1u);
    if (pos < (u32)TOPK) { topKey[lvl * TOPK + pos] = key; topIdx[lvl * TOPK + pos] = idxL; }
  } else {
    u32 pos = atomicAdd(&meta[M_BND + lvl], 1u);
    if (pos < (u32)BND_CAP)
      bnd[(lvl << 13) + pos] = ((u64)key << 32) | (u64)(0xFFFFFFFFu - idxL);
  }
}

__global__ void __launch_bounds__(256) k_compact(const float* __restrict__ logits, u32* meta,
                                                 u32* topKey, u32* topIdx, u64* bnd) {
  u32 cuts[5];
#pragma unroll
  for (int l = 0; l < 5; ++l) cuts[l] = meta[M_CUT + l];
  int gid = blockIdx.x * blockDim.x + threadIdx.x;
  int stride = gridDim.x * blockDim.x;
  const int n4 = SCORE_TOTAL >> 2;
  for (int i4 = gid; i4 < n4; i4 += stride) {
    float4 v = ((const float4*)logits)[i4];
    int b = i4 * 4;
    compactOne(b + 0, v.x, cuts, meta, topKey, topIdx, bnd);
    compactOne(b + 1, v.y, cuts, meta, topKey, topIdx, bnd);
    compactOne(b + 2, v.z, cuts, meta, topKey, topIdx, bnd);
    compactOne(b + 3, v.w, cuts, meta, topKey, topIdx, bnd);
  }
  for (int i = n4 * 4 + gid; i < SCORE_TOTAL; i += stride)
    compactOne(i, logits[i], cuts, meta, topKey, topIdx, bnd);
}

// ---------------- K4: boundary-bin selection (per-level bitonic) ----------------
__global__ void __launch_bounds__(1024) k_boundary(const u64* __restrict__ bnd, const u32* __restrict__ meta,
                                                   u32* topKey, u32* topIdx) {
  __shared__ u64 arr[BND_CAP];
  int l = blockIdx.x, tid = threadIdx.x;
  u32 cnt = min(meta[M_BND + l], (u32)BND_CAP);
  for (int i = tid; i < BND_CAP; i += 1024) arr[i] = (i < (int)cnt) ? bnd[(l << 13) + i] : 0ull;
  __syncthreads();
  bitonicDesc(arr, BND_CAP, tid, 1024);
  u32 need = meta[M_NEED + l], above = meta[M_ABOVE + l];
  for (u32 t = tid; t < need; t += 1024) {
    u64 pk = arr[t];
    u32 key = (u32)(pk >> 32);
    u32 idx = 0xFFFFFFFFu - (u32)pk;
    if (pk == 0ull) { key = 0u; idx = 0u; }
    topKey[l * TOPK + above + t] = key;
    topIdx[l * TOPK + above + t] = idx;
  }
}

// ---------------- K5: decode boxes, build candidates ----------------
__global__ void k_decode(const float* __restrict__ breg, const float* __restrict__ anc,
                         const u32* __restrict__ topKey, const u32* __restrict__ topIdx,
                         float4* candBoff, float4* candBox, float* candScore, u32* candLabel) {
  int c = blockIdx.x * 256 + threadIdx.x;
  if (c >= NCAND_PAD) return;
  float4 pad = make_float4(1e7f, 1e7f, 1e7f, 1e7f); // zero-area, zero-IoU sentinel
  if (c >= NCAND || topKey[c] == 0u) {
    candBoff[c] = pad; candBox[c] = make_float4(0.f, 0.f, 0.f, 0.f);
    candScore[c] = -1.0f; candLabel[c] = 0u;
    return;
  }
  int l = c / TOPK;
  u32 key = topKey[c];
  u32 idx = topIdx[c];
  int aI = (int)(idx / 91u);
  int cls = (int)(idx - (u32)aI * 91u);
  int g = anchOff(l) + aI;
  float logit = keyToFloat(key);
  float score = 1.0f / (1.0f + expf(-logit));
  float ax1 = anc[g * 4 + 0], ay1 = anc[g * 4 + 1], ax2 = anc[g * 4 + 2], ay2 = anc[g * 4 + 3];
  float w = ax2 - ax1, h = ay2 - ay1;
  float cx = ax1 + 0.5f * w, cy = ay1 + 0.5f * h;
  float dx = breg[g * 4 + 0], dy = breg[g * 4 + 1];
  float dw = fminf(breg[g * 4 + 2], BBOX_CLIPF), dh = fminf(breg[g * 4 + 3], BBOX_CLIPF);
  float pcx = dx * w + cx, pcy = dy * h + cy;
  float pw = expf(dw) * w, ph = expf(dh) * h;
  float x1 = fminf(fmaxf(pcx - 0.5f * pw, 0.f), IMGF);
  float y1 = fminf(fmaxf(pcy - 0.5f * ph, 0.f), IMGF);
  float x2 = fminf(fmaxf(pcx + 0.5f * pw, 0.f), IMGF);
  float y2 = fminf(fmaxf(pcy + 0.5f * ph, 0.f), IMGF);
  float off = (float)cls * (IMGF + 1.0f);
  candBox[c] = make_float4(x1, y1, x2, y2);
  candBoff[c] = make_float4(x1 + off, y1 + off, x2 + off, y2 + off);
  candScore[c] = score;
  candLabel[c] = (u32)cls;
}

// ---------------- K6: global sort by score (desc) ----------------
__global__ void __launch_bounds__(1024) k_sort(const float* __restrict__ candScore, u32* order) {
  __shared__ u64 arr[8192];
  int tid = threadIdx.x;
  for (int i = tid; i < 8192; i += 1024)
    arr[i] = (i < NCAND) ? (((u64)orderKey(candScore[i]) << 32) | (u64)(0xFFFFFFFFu - (u32)i)) : 0ull;
  __syncthreads();
  bitonicDesc(arr, 8192, tid, 1024);
  for (int i = tid; i < NCAND_PAD; i += 1024)
    order[i] = (i < NCAND) ? (0xFFFFFFFFu - (u32)arr[i]) : (u32)i;
}

// ---------------- K6b: gather sorted candidate arrays ----------------
__global__ void k_gather(const u32* __restrict__ order,
                         const float4* __restrict__ cBoff, const float4* __restrict__ cBox,
                         const float* __restrict__ cS, const u32* __restrict__ cL,
                         float4* sBoff, float4* sBox, float* sS, u32* sL, float* sA) {
  int i = blockIdx.x * 256 + threadIdx.x;
  if (i >= NCAND_PAD) return;
  u32 o = order[i];
  float4 bf = cBoff[o];
  sBoff[i] = bf; sBox[i] = cBox[o]; sS[i] = cS[o]; sL[i] = cL[o];
  sA[i] = fmaxf(bf.z - bf.x, 0.f) * fmaxf(bf.w - bf.y, 0.f);
}

// ---------------- K7: IoU suppression bitmask (WMMA union + wave32 ballot) ----------------
__global__ void __launch_bounds__(256) k_iou(const float4* __restrict__ sBoff,
                                             const float* __restrict__ sArea,
                                             u32* __restrict__ mask) {
  int wave = (blockIdx.x * blockDim.x + threadIdx.x) >> 5;
  int lane = threadIdx.x & 31;
  if (wave >= NTI * NJW) return;
  int ti = wave / NJW, tw = wave - ti * NJW;
  int i0 = ti * 16, j0 = tw * 32;
  int lc = lane & 15;
  int hi = lane >> 4;

  float4 rB = sBoff[i0 + lc];        // row box owned by lane lc (dup across halves)
  float  rA = sArea[i0 + lc];
  float4 cB0 = sBoff[j0 + lc];       // this lane's column boxes (both wave halves use cols 0..15)
  float4 cB1 = sBoff[j0 + 16 + lc];
  float  cA0 = sArea[j0 + lc];
  float  cA1 = sArea[j0 + 16 + lc];

#if HAVE_WMMA4
  // A(16x4): row M -> [area_M, 1, 0, 0]; B(4x16): col N -> [1, area_N, 0, 0]^T
  // lanes 0-15 hold K=0,1 (A) / K=0,1 (B); lanes 16-31 hold K=2,3 -> zeros
  v2f a, b0v, b1v;
  a.x = hi ? 0.f : rA;   a.y = hi ? 0.f : 1.f;
  b0v.x = hi ? 0.f : 1.f; b0v.y = hi ? 0.f : cA0;
  b1v.x = hi ? 0.f : 1.f; b1v.y = hi ? 0.f : cA1;
  v8f cz = {0.f, 0.f, 0.f, 0.f, 0.f, 0.f, 0.f, 0.f};
  v8f d0 = __builtin_amdgcn_wmma_f32_16x16x4_f32(false, a, false, b0v, (short)0, cz, false, false);
  v8f d1 = __builtin_amdgcn_wmma_f32_16x16x4_f32(false, a, false, b1v, (short)0, cz, false, false);
#endif

  u32 bal0[8], bal1[8];
#pragma unroll
  for (int r = 0; r < 8; ++r) {
    int src = r + 8 * hi;            // this lane's row within the 16-row tile
    float rx1 = __shfl(rB.x, src, 32);
    float ry1 = __shfl(rB.y, src, 32);
    float rx2 = __shfl(rB.z, src, 32);
    float ry2 = __shfl(rB.w, src, 32);
#if HAVE_WMMA4
    float sum0 = d0[r], sum1 = d1[r];
#else
    float rAr = __shfl(rA, src, 32);
    float sum0 = rAr + cA0, sum1 = rAr + cA1;
#endif
    float iw0 = fmaxf(fminf(rx2, cB0.z) - fmaxf(rx1, cB0.x), 0.f);
    float ih0 = fmaxf(fminf(ry2, cB0.w) - fmaxf(ry1, cB0.y), 0.f);
    float in0 = iw0 * ih0;
    float iou0 = in0 / fmaxf(sum0 - in0, 1e-9f);
    bal0[r] = ballot32(iou0 > 0.5f);

    float iw1 = fmaxf(fminf(rx2, cB1.z) - fmaxf(rx1, cB1.x), 0.f);
    float ih1 = fmaxf(fminf(ry2, cB1.w) - fmaxf(ry1, cB1.y), 0.f);
    float in1 = iw1 * ih1;
    float iou1 = in1 / fmaxf(sum1 - in1, 1e-9f);
    bal1[r] = ballot32(iou1 > 0.5f);
  }
  // ballots are wave-uniform; lanes 0..15 each write one row's 32-bit word
  if (lane < 16) {
    int r = lane & 7, h2 = lane >> 3;
    u32 w = h2 ? ((bal0[r] >> 16) | (bal1[r] & 0xFFFF0000u))
               : ((bal0[r] & 0xFFFFu) | (bal1[r] << 16));
    mask[(i0 + r + 8 * h2) * MASK_STRIDE + tw] = w;
  }
}

// ---------------- K8: single-wave greedy NMS scan ----------------
__global__ void __launch_bounds__(32) k_nms(const u32* __restrict__ mask, u32* keepW) {
  int lane = threadIdx.x;                   // one wave32; lane owns words {lane, lane+32, ...}
  u32 rem[5] = {0, 0, 0, 0, 0};
  u32 kp[5] = {0, 0, 0, 0, 0};
  for (int i = 0; i < NCAND; ++i) {
    int w = i >> 5;
    int slot = i >> 10;                     // uniform across wave
    int owner = w & 31;
    u32 fw = __shfl(rem[slot], owner, 32);  // broadcast removed-word for i
    if (!((fw >> (i & 31)) & 1u)) {
      if (lane == owner) kp[slot] |= (1u << (i & 31));
      const u32* row = mask + i * MASK_STRIDE;
#pragma unroll
      for (int s = 0; s < 5; ++s) {
        int wi = s * 32 + lane;
        if (wi < NJW) rem[s] |= row[wi];
      }
    }
  }
#pragma unroll
  for (int s = 0; s < 5; ++s) {
    int wi = s * 32 + lane;
    if (wi < MASK_STRIDE) keepW[wi] = (wi < NJW) ? kp[s] : 0u;
  }
}

// ---------------- K9: emit final 300 detections ----------------
__global__ void __launch_bounds__(1024) k_final(const float4* __restrict__ sBox,
                                                const float* __restrict__ sScore,
                                                const u32* __restrict__ sLabel,
                                                const u32* __restrict__ keepW,
                                                float* __restrict__ out) {
  __shared__ int sc[1024];
  int tid = threadIdx.x;
  int base = 0;
  for (int c = 0; c < 5; ++c) {
    int i = c * 1024 + tid;
    int valid = 0;
    float score = 0.f;
    if (i < NCAND) {
      score = sScore[i];
      valid = (((keepW[i >> 5] >> (i & 31)) & 1u) && (score > 0.f)) ? 1 : 0;
    }
    sc[tid] = valid;
    __syncthreads();
    for (int off = 1; off < 1024; off <<= 1) {
      int v = (tid >= off) ? sc[tid - off] : 0;
      __syncthreads();
      sc[tid] += v;
      __syncthreads();
    }
    int rank = base + sc[tid] - valid;
    int total = sc[1023];
    if (valid && rank < DETS) {
      float4 b = sBox[i];
      out[rank * 6 + 0] = b.x; out[rank * 6 + 1] = b.y;
      out[rank * 6 + 2] = b.z; out[rank * 6 + 3] = b.w;
      out[rank * 6 + 4] = score;
      out[rank * 6 + 5] = (float)sLabel[i];
    }
    base += total;
    __syncthreads();
  }
  int nv = base < DETS ? base : DETS;
  for (int s = nv + tid; s < DETS; s += 1024) {
    out[s * 6 + 0] = 0.f; out[s * 6 + 1] = 0.f; out[s * 6 + 2] = 0.f; out[s * 6 + 3] = 0.f;
    out[s * 6 + 4] = 0.f; out[s * 6 + 5] = -1.0f;
  }
}

// ---------------- host launcher ----------------
extern "C" void kernel_launch(void* const* d_in, const int* in_sizes, int n_in,
                              void* d_out, int out_size, void* d_ws, size_t ws_size,
                              hipStream_t stream) {
  (void)in_sizes; (void)n_in; (void)out_size; (void)ws_size;
  const float* logits = (const float*)d_in[0];
  const float* breg = (const float*)d_in[1];
  const float* anch = (const float*)d_in[2];
  float* out = (float*)d_out;

  char* p = (char*)d_ws;
  auto alloc = [&](size_t b) { char* r = p; p += (b + 255) & ~(size_t)255; return r; };
  u32* hist = (u32*)alloc(5120 * sizeof(u32));
  u32* meta = (u32*)alloc(64 * sizeof(u32));
  u32* topKey = (u32*)alloc(NCAND * sizeof(u32));
  u32* topIdx = (u32*)alloc(NCAND * sizeof(u32));
  u64* bnd = (u64*)alloc((size_t)5 * BND_CAP * sizeof(u64));
  float4* candBoff = (float4*)alloc(NCAND_PAD * sizeof(float4));
  float4* candBox = (float4*)alloc(NCAND_PAD * sizeof(float4));
  float* candScore = (float*)alloc(NCAND_PAD * sizeof(float));
  u32* candLabel = (u32*)alloc(NCAND_PAD * sizeof(u32));
  u32* order = (u32*)alloc(NCAND_PAD * sizeof(u32));
  float4* sBoff = (float4*)alloc(NCAND_PAD * sizeof(float4));
  float4* sBox = (float4*)alloc(NCAND_PAD * sizeof(float4));
  float* sScore = (float*)alloc(NCAND_PAD * sizeof(float));
  u32* sLabel = (u32*)alloc(NCAND_PAD * sizeof(u32));
  float* sArea = (float*)alloc(NCAND_PAD * sizeof(float));
  u32* mask = (u32*)alloc((size_t)NCAND_PAD * MASK_STRIDE * sizeof(u32));
  u32* keepW = (u32*)alloc(MASK_STRIDE * sizeof(u32));

  k_init<<<1, 1024, 0, stream>>>(hist, meta);
  k_hist<<<668, 256, 0, stream>>>(logits, hist);
  k_cutoff<<<1, 64, 0, stream>>>(hist, meta);
  k_compact<<<1024, 256, 0, stream>>>(logits, meta, topKey, topIdx, bnd);
  k_boundary<<<5, 1024, 0, stream>>>(bnd, meta, topKey, topIdx);
  k_decode<<<(NCAND_PAD + 255) / 256, 256, 0, stream>>>(breg, anch, topKey, topIdx,
                                                        candBoff, candBox, candScore, candLabel);
  k_sort<<<1, 1024, 0, stream>>>(candScore, order);
  k_gather<<<(NCAND_PAD + 255) / 256, 256, 0, stream>>>(order, candBoff, candBox, candScore, candLabel,
                                                        sBoff, sBox, sScore, sLabel, sArea);
  k_iou<<<(NTI * NJW + 7) / 8, 256, 0, stream>>>(sBoff, sArea, mask);
  k_nms<<<1, 32, 0, stream>>>(mask, keepW);
  k_final<<<1, 1024, 0, stream>>>(sBox, sScore, sLabel, keepW, out);
}